// ConstraintGraphSAGE_51376398795504
// MI455X (gfx1250) — compile-verified
//
#include <hip/hip_runtime.h>
#include <hip/hip_bf16.h>
#include <cstdint>

#define LN_EPS 1e-5f

typedef float v2f __attribute__((ext_vector_type(2)));
typedef float v8f __attribute__((ext_vector_type(8)));

// ---------- helpers ----------
__device__ __forceinline__ unsigned fkey(float f) {
  unsigned b = __float_as_uint(f);
  return (b & 0x80000000u) ? ~b : (b | 0x80000000u);   // order-preserving uint key
}
__device__ __forceinline__ float funkey(unsigned k) {
  unsigned b = (k & 0x80000000u) ? (k ^ 0x80000000u) : ~k;
  return __uint_as_float(b);
}
__device__ __forceinline__ float lrelu(float v) { return v > 0.f ? v : 0.2f * v; }

// ---------- edge_attr column sums (mean computed as sum * invE) ----------
__global__ void edge_mean_sum_kernel(const float* __restrict__ ea,
                                     float* __restrict__ sums, int E) {
  float acc[5] = {0.f, 0.f, 0.f, 0.f, 0.f};
  const int stride = gridDim.x * blockDim.x;
  for (int e = blockIdx.x * blockDim.x + threadIdx.x; e < E; e += stride) {
#pragma unroll
    for (int j = 0; j < 5; ++j) acc[j] += ea[(size_t)e * 5 + j];
  }
#pragma unroll
  for (int j = 0; j < 5; ++j) atomicAdd(&sums[j], acc[j]);
}

// ---------- fp32 WMMA GEMM: C[M x OC] = A[M x K] @ W[K x OC] + bias ----------
// Compile-time K/OC: constant-folded addressing, K chunked by 16 so each chunk
// issues a clause of loads followed by 4 back-to-back V_WMMA_F32_16X16X4_F32.
// Requires M % 16 == 0 (true: M = N = 100000 = 6250*16).
template <int K, int OC>
__global__ void gemm_wmma_kernel(const float* __restrict__ A, const float* __restrict__ W,
                                 const float* __restrict__ bias, float* __restrict__ C,
                                 int M) {
  constexpr int TN = OC / 16;
  const int lane = threadIdx.x & 31;
  const int wave = threadIdx.x >> 5;
  const int tile = blockIdx.x * (blockDim.x >> 5) + wave;
  const int tiles_m = M >> 4;
  if (tile >= tiles_m * TN) return;               // wave-uniform: EXEC untouched
  const int tm = tile / TN, tn = tile % TN;
  const int half = lane >> 4;                     // selects K-pair (A/B) and M-half (C/D)
  const int l16  = lane & 15;
  // A fragment base: lane = M row, vgpr pair at K = k + half*2
  const float* arow = A + (size_t)(tm * 16 + l16) * K + half * 2;
  // B fragment base: lane = N col, vgpr pair at K rows (k + half*2), (k + half*2 + 1)
  const float* bcol = W + (size_t)(half * 2) * OC + tn * 16 + l16;
  v8f acc = {};
#pragma unroll
  for (int k0 = 0; k0 < K; k0 += 16) {
    v2f a[4], b[4];
#pragma unroll
    for (int i = 0; i < 4; ++i) {                 // load clause: 4x b64 + 8x b32
      const int k = k0 + 4 * i;
      a[i] = *(const v2f*)(arow + k);
      b[i].x = bcol[(size_t)k * OC];
      b[i].y = bcol[(size_t)(k + 1) * OC];
    }
#pragma unroll
    for (int i = 0; i < 4; ++i)                   // 4 back-to-back WMMAs
      acc = __builtin_amdgcn_wmma_f32_16x16x4_f32(false, a[i], false, b[i],
                                                  (short)0, acc, false, false);
  }
  const float bv = bias[tn * 16 + l16];
#pragma unroll
  for (int r = 0; r < 8; ++r) {                   // C/D: vgpr r -> M = r + half*8
    const int row = tm * 16 + half * 8 + r;
    C[(size_t)row * OC + tn * 16 + l16] = acc[r] + bv;
  }
}

// ---------- edge pass A: logits + segment max (one wave per edge) ----------
template <int F, int H>
__global__ void edge_logits_kernel(const float* __restrict__ xl, const float* __restrict__ xr,
                                   const int* __restrict__ ei, const float* __restrict__ eattr,
                                   const float* __restrict__ msum, float invE,
                                   const float* __restrict__ We, const float* __restrict__ att,
                                   float* __restrict__ elog, unsigned* __restrict__ lmax,
                                   int N, int E) {
  constexpr int PF  = F / 32;   // features per lane
  constexpr int C   = F / H;    // channels per head
  constexpr int LPH = 32 / H;   // lanes per head
  const int lane = threadIdx.x & 31;
  const int e = blockIdx.x * (blockDim.x >> 5) + (threadIdx.x >> 5);
  const int E2 = E + N;
  if (e >= E2) return;
  int s, d;
  if (e < E) { s = ei[e]; d = ei[E + e]; } else { s = e - E; d = s; }
  float ea[5];
#pragma unroll
  for (int j = 0; j < 5; ++j)
    ea[j] = (e < E) ? eattr[(size_t)e * 5 + j] : msum[j] * invE;  // self-loop: mean attr
  const int f0 = lane * PF;
  const int h  = lane / LPH;
  const float* xlp = xl + (size_t)s * F + f0;
  const float* xrp = xr + (size_t)d * F + f0;
  float p = 0.f;
#pragma unroll
  for (int t = 0; t < PF; ++t) {
    const int f = f0 + t;
    float v = xlp[t] + xrp[t];
#pragma unroll
    for (int j = 0; j < 5; ++j) v += ea[j] * We[j * F + f];       // ee recomputed on the fly
    p += att[h * C + (f - h * C)] * lrelu(v);
  }
#pragma unroll
  for (int m = LPH / 2; m >= 1; m >>= 1) p += __shfl_xor(p, m, 32);
  if ((lane & (LPH - 1)) == 0) {
    elog[(size_t)e * H + h] = p;
    atomicMax(&lmax[(size_t)d * H + h], fkey(p));
  }
}

// ---------- edge pass B: ex = exp(l - lmax[dst]); denom[dst] += ex ----------
template <int H>
__global__ void edge_softmax_kernel(float* __restrict__ elog, const unsigned* __restrict__ lmax,
                                    float* __restrict__ denom, const int* __restrict__ ei,
                                    int N, int E) {
  const int e = blockIdx.x * blockDim.x + threadIdx.x;
  const int E2 = E + N;
  if (e >= E2) return;
  const int d = (e < E) ? ei[E + e] : (e - E);
#pragma unroll
  for (int h = 0; h < H; ++h) {
    const float lm = funkey(lmax[(size_t)d * H + h]);   // every dst has a self-loop -> finite
    const float ex = __expf(elog[(size_t)e * H + h] - lm);
    elog[(size_t)e * H + h] = ex;
    atomicAdd(&denom[(size_t)d * H + h], ex);
  }
}

// ---------- edge pass C: agg[dst] += xl[src] * alpha (one wave per edge) ----------
template <int F, int H>
__global__ void edge_scatter_kernel(const float* __restrict__ elog, const float* __restrict__ denom,
                                    const float* __restrict__ xl, const int* __restrict__ ei,
                                    float* __restrict__ agg, int N, int E) {
  constexpr int PF  = F / 32;
  constexpr int LPH = 32 / H;
  const int lane = threadIdx.x & 31;
  const int e = blockIdx.x * (blockDim.x >> 5) + (threadIdx.x >> 5);
  const int E2 = E + N;
  if (e >= E2) return;
  int s, d;
  if (e < E) { s = ei[e]; d = ei[E + e]; } else { s = e - E; d = s; }
  const int h = lane / LPH;
  const float alpha = elog[(size_t)e * H + h] / (denom[(size_t)d * H + h] + 1e-16f);
  const int f0 = lane * PF;
  const float* xlp = xl + (size_t)s * F + f0;
  float* ap = agg + (size_t)d * F + f0;
#pragma unroll
  for (int t = 0; t < PF; ++t)
    atomicAdd(&ap[t], xlp[t] * alpha);
}

// ---------- bias + LayerNorm + ReLU (+skip) (+L2 normalize) : one wave per node ----------
template <int F, bool ADDSKIP, bool L2N>
__global__ void ln_relu_kernel(const float* __restrict__ in, const float* __restrict__ bias,
                               const float* __restrict__ g, const float* __restrict__ b,
                               const float* __restrict__ skip, float* __restrict__ out, int N) {
  constexpr int PF = F / 32;
  const int lane = threadIdx.x & 31;
  const int n = blockIdx.x * (blockDim.x >> 5) + (threadIdx.x >> 5);
  if (n >= N) return;
  float v[PF];
  float s = 0.f;
#pragma unroll
  for (int t = 0; t < PF; ++t) {
    v[t] = in[(size_t)n * F + lane * PF + t] + bias[lane * PF + t];
    s += v[t];
  }
#pragma unroll
  for (int m = 16; m >= 1; m >>= 1) s += __shfl_xor(s, m, 32);
  const float mu = s * (1.0f / F);
  float q = 0.f;
#pragma unroll
  for (int t = 0; t < PF; ++t) { const float dlt = v[t] - mu; q += dlt * dlt; }
#pragma unroll
  for (int m = 16; m >= 1; m >>= 1) q += __shfl_xor(q, m, 32);
  const float rstd = rsqrtf(q * (1.0f / F) + LN_EPS);
  float o[PF];
  float nrm = 0.f;
#pragma unroll
  for (int t = 0; t < PF; ++t) {
    const int f = lane * PF + t;
    float y = (v[t] - mu) * rstd * g[f] + b[f];
    y = fmaxf(y, 0.f);                               // ReLU
    if (ADDSKIP) y += skip[(size_t)n * F + f];       // skip added AFTER relu (per reference)
    o[t] = y;
    nrm += y * y;
  }
  if (L2N) {
#pragma unroll
    for (int m = 16; m >= 1; m >>= 1) nrm += __shfl_xor(nrm, m, 32);
    const float inv = 1.0f / fmaxf(sqrtf(nrm), 1e-12f);
#pragma unroll
    for (int t = 0; t < PF; ++t) o[t] *= inv;
  }
#pragma unroll
  for (int t = 0; t < PF; ++t) out[(size_t)n * F + lane * PF + t] = o[t];
}

// ---------- host orchestration ----------
extern "C" void kernel_launch(void* const* d_in, const int* in_sizes, int n_in,
                              void* d_out, int out_size, void* d_ws, size_t ws_size,
                              hipStream_t stream) {
  const float* x      = (const float*)d_in[0];
  const int*   ei     = (const int*)d_in[1];
  const float* eattr  = (const float*)d_in[2];
  const float* skip_W = (const float*)d_in[3];
  const float* skip_b = (const float*)d_in[4];
  const float* Wl1 = (const float*)d_in[5];  const float* bl1 = (const float*)d_in[6];
  const float* Wr1 = (const float*)d_in[7];  const float* br1 = (const float*)d_in[8];
  const float* We1 = (const float*)d_in[9];  const float* att1 = (const float*)d_in[10];
  const float* bias1 = (const float*)d_in[11];
  const float* g1 = (const float*)d_in[12];  const float* b1 = (const float*)d_in[13];
  const float* Wl2 = (const float*)d_in[14]; const float* bl2 = (const float*)d_in[15];
  const float* Wr2 = (const float*)d_in[16]; const float* br2 = (const float*)d_in[17];
  const float* We2 = (const float*)d_in[18]; const float* att2 = (const float*)d_in[19];
  const float* bias2 = (const float*)d_in[20];
  const float* g2 = (const float*)d_in[21];  const float* b2 = (const float*)d_in[22];
  const float* Wl3 = (const float*)d_in[23]; const float* bl3 = (const float*)d_in[24];
  const float* Wr3 = (const float*)d_in[25]; const float* br3 = (const float*)d_in[26];
  const float* We3 = (const float*)d_in[27]; const float* att3 = (const float*)d_in[28];
  const float* bias3 = (const float*)d_in[29];
  const float* g3 = (const float*)d_in[30];  const float* b3 = (const float*)d_in[31];

  const int N  = in_sizes[0] / 16;
  const int E  = in_sizes[2] / 5;
  const int E2 = E + N;
  const float invE = 1.0f / (float)E;

  // workspace carve-out
  float* ws = (float*)d_ws;
  size_t off = 0;
  float*    msum  = ws + off; off += 8;
  float*    skip  = ws + off; off += (size_t)N * 128;
  float*    xl    = ws + off; off += (size_t)N * 128;
  float*    xr    = ws + off; off += (size_t)N * 128;
  float*    hA    = ws + off; off += (size_t)N * 128;
  float*    agg   = ws + off; off += (size_t)N * 128;
  float*    elog  = ws + off; off += (size_t)E2 * 4;     // logits then ex, in place
  unsigned* lmax  = (unsigned*)(ws + off); off += (size_t)N * 4;
  float*    denom = ws + off; off += (size_t)N * 4;

  // 256-thread blocks = 8 waves; one 16x16 output tile per wave
  const int gb128_16  = ((N / 16) * 8 + 7) / 8;   // K=16 , OC=128
  const int gb128_128 = ((N / 16) * 8 + 7) / 8;   // K=128, OC=128
  const int gb64_128  = ((N / 16) * 4 + 7) / 8;   // K=128, OC=64

  const int eb = (E2 + 7) / 8;      // wave-per-edge, 8 waves per 256-thread block
  const int tb = (E2 + 255) / 256;  // thread-per-edge
  const int nb = (N + 7) / 8;       // wave-per-node

  // edge_attr mean (as sums; consumers scale by invE)
  hipMemsetAsync(msum, 0, 8 * sizeof(float), stream);
  edge_mean_sum_kernel<<<256, 256, 0, stream>>>(eattr, msum, E);

  // skip branch
  gemm_wmma_kernel<16, 128><<<gb128_16, 256, 0, stream>>>(x, skip_W, skip_b, skip, N);

  // ---------------- layer 1 (16 -> 4x32 concat) ----------------
  gemm_wmma_kernel<16, 128><<<gb128_16, 256, 0, stream>>>(x, Wl1, bl1, xl, N);
  gemm_wmma_kernel<16, 128><<<gb128_16, 256, 0, stream>>>(x, Wr1, br1, xr, N);
  hipMemsetAsync(lmax,  0, (size_t)N * 4 * sizeof(unsigned), stream);
  hipMemsetAsync(denom, 0, (size_t)N * 4 * sizeof(float), stream);
  hipMemsetAsync(agg,   0, (size_t)N * 128 * sizeof(float), stream);
  edge_logits_kernel<128, 4><<<eb, 256, 0, stream>>>(xl, xr, ei, eattr, msum, invE,
                                                     We1, att1, elog, lmax, N, E);
  edge_softmax_kernel<4><<<tb, 256, 0, stream>>>(elog, lmax, denom, ei, N, E);
  edge_scatter_kernel<128, 4><<<eb, 256, 0, stream>>>(elog, denom, xl, ei, agg, N, E);
  ln_relu_kernel<128, true, false><<<nb, 256, 0, stream>>>(agg, bias1, g1, b1, skip, hA, N);

  // ---------------- layer 2 (128 -> 4x32 concat) ----------------
  gemm_wmma_kernel<128, 128><<<gb128_128, 256, 0, stream>>>(hA, Wl2, bl2, xl, N);
  gemm_wmma_kernel<128, 128><<<gb128_128, 256, 0, stream>>>(hA, Wr2, br2, xr, N);
  hipMemsetAsync(lmax,  0, (size_t)N * 4 * sizeof(unsigned), stream);
  hipMemsetAsync(denom, 0, (size_t)N * 4 * sizeof(float), stream);
  hipMemsetAsync(agg,   0, (size_t)N * 128 * sizeof(float), stream);
  edge_logits_kernel<128, 4><<<eb, 256, 0, stream>>>(xl, xr, ei, eattr, msum, invE,
                                                     We2, att2, elog, lmax, N, E);
  edge_softmax_kernel<4><<<tb, 256, 0, stream>>>(elog, lmax, denom, ei, N, E);
  edge_scatter_kernel<128, 4><<<eb, 256, 0, stream>>>(elog, denom, xl, ei, agg, N, E);
  ln_relu_kernel<128, false, false><<<nb, 256, 0, stream>>>(agg, bias2, g2, b2, nullptr, hA, N);

  // ---------------- layer 3 (128 -> 1x64, mean == identity for H=1) ----------------
  gemm_wmma_kernel<128, 64><<<gb64_128, 256, 0, stream>>>(hA, Wl3, bl3, xl, N);
  gemm_wmma_kernel<128, 64><<<gb64_128, 256, 0, stream>>>(hA, Wr3, br3, xr, N);
  hipMemsetAsync(lmax,  0, (size_t)N * sizeof(unsigned), stream);
  hipMemsetAsync(denom, 0, (size_t)N * sizeof(float), stream);
  hipMemsetAsync(agg,   0, (size_t)N * 64 * sizeof(float), stream);
  edge_logits_kernel<64, 1><<<eb, 256, 0, stream>>>(xl, xr, ei, eattr, msum, invE,
                                                    We3, att3, elog, lmax, N, E);
  edge_softmax_kernel<1><<<tb, 256, 0, stream>>>(elog, lmax, denom, ei, N, E);
  edge_scatter_kernel<64, 1><<<eb, 256, 0, stream>>>(elog, denom, xl, ei, agg, N, E);
  ln_relu_kernel<64, false, true><<<nb, 256, 0, stream>>>(agg, bias3, g3, b3, nullptr,
                                                          (float*)d_out, N);
}